// FactorizedPrior_64707977281655
// MI455X (gfx1250) — compile-verified
//
#include <hip/hip_runtime.h>
#include <stdint.h>

#define CC      192
#define HW      4096            // 64*64
#define NSL     (32 * 192)      // 6144 slices (B*C)
#define NTOT    (NSL * HW)      // 25165824 elements
#define PSTRIDE 48              // floats per channel param record
#define NT      4               // tiles per slice; tile = 1024 floats = 4KB

typedef float v2f __attribute__((ext_vector_type(2)));
typedef float v4f __attribute__((ext_vector_type(4)));
typedef float v8f __attribute__((ext_vector_type(8)));

// ---------------- fast device math ----------------
__device__ __forceinline__ float fast_exp2(float x) {
#if __has_builtin(__builtin_amdgcn_exp2f)
  return __builtin_amdgcn_exp2f(x);        // v_exp_f32
#else
  return exp2f(x);
#endif
}
__device__ __forceinline__ float fast_rcp(float x) {
#if __has_builtin(__builtin_amdgcn_rcpf)
  return __builtin_amdgcn_rcpf(x);         // v_rcp_f32
#else
  return 1.0f / x;
#endif
}
__device__ __forceinline__ float fast_tanh(float x) {
  // tanh(x) = 1 - 2/(1 + 2^(2*log2(e)*x)); saturates correctly at +/-1
  float e = fast_exp2(x * 2.885390081777927f);
  return 1.0f - 2.0f * fast_rcp(e + 1.0f);
}
__device__ __forceinline__ float fast_sigmoid(float x) {
  return fast_rcp(1.0f + fast_exp2(-1.4426950408889634f * x));
}
__device__ __forceinline__ float softplus_f(float x) {
  return fmaxf(x, 0.0f) + log1pf(expf(-fabsf(x)));
}

// ---------------- per-channel parameter precompute (tiny) ----------------
// Record layout (floats, stride 48):
// [0:3)   w0          [3:6)   b0          [6:15)  W1 (o*3+i)  [15:18) b1
// [18:27) W2 (o*3+i)  [27:30) b2          [30:33) w3          [33]    b3
// [34:37) tanh(f0)    [37:40) tanh(f1)    [40:43) tanh(f2)
// [43] flag (1.0 if all f == 0)   [44] a (collapsed gain)   [45] d (collapsed bias)
__global__ void fp_prep(const float* __restrict__ m0, const float* __restrict__ m1,
                        const float* __restrict__ m2, const float* __restrict__ m3,
                        const float* __restrict__ b0, const float* __restrict__ b1,
                        const float* __restrict__ b2, const float* __restrict__ b3,
                        const float* __restrict__ f0, const float* __restrict__ f1,
                        const float* __restrict__ f2, float* __restrict__ P) {
  int c = blockIdx.x * blockDim.x + threadIdx.x;
  if (c >= CC) return;
  float* o = P + c * PSTRIDE;

  float w0[3], B0[3], W1[9], B1[3], W2[9], B2[3], w3[3];
  float B3 = b3[c];
  bool allz = true;
#pragma unroll
  for (int j = 0; j < 3; ++j) {
    w0[j] = softplus_f(m0[c * 3 + j]);
    w3[j] = softplus_f(m3[c * 3 + j]);
    B0[j] = b0[c * 3 + j];
    B1[j] = b1[c * 3 + j];
    B2[j] = b2[c * 3 + j];
    float F0 = f0[c * 3 + j], F1 = f1[c * 3 + j], F2 = f2[c * 3 + j];
    allz = allz && (F0 == 0.0f) && (F1 == 0.0f) && (F2 == 0.0f);
    o[34 + j] = tanhf(F0);
    o[37 + j] = tanhf(F1);
    o[40 + j] = tanhf(F2);
  }
#pragma unroll
  for (int k = 0; k < 9; ++k) {
    W1[k] = softplus_f(m1[c * 9 + k]);
    W2[k] = softplus_f(m2[c * 9 + k]);
  }
#pragma unroll
  for (int j = 0; j < 3; ++j) { o[0 + j] = w0[j]; o[3 + j] = B0[j];
                                o[15 + j] = B1[j]; o[27 + j] = B2[j];
                                o[30 + j] = w3[j]; }
#pragma unroll
  for (int k = 0; k < 9; ++k) { o[6 + k] = W1[k]; o[18 + k] = W2[k]; }
  o[33] = B3;

  // Collapse to affine: logit(x) = a*x + d   (exact when all factors are 0)
  float r3[3] = {w3[0], w3[1], w3[2]};
  float dacc = B3 + r3[0] * B2[0] + r3[1] * B2[1] + r3[2] * B2[2];
  float r2[3], r1[3];
#pragma unroll
  for (int j = 0; j < 3; ++j)
    r2[j] = r3[0] * W2[0 * 3 + j] + r3[1] * W2[1 * 3 + j] + r3[2] * W2[2 * 3 + j];
  dacc += r2[0] * B1[0] + r2[1] * B1[1] + r2[2] * B1[2];
#pragma unroll
  for (int j = 0; j < 3; ++j)
    r1[j] = r2[0] * W1[0 * 3 + j] + r2[1] * W1[1 * 3 + j] + r2[2] * W1[2 * 3 + j];
  dacc += r1[0] * B0[0] + r1[1] * B0[1] + r1[2] * B0[2];
  float aacc = r1[0] * w0[0] + r1[1] * w0[1] + r1[2] * w0[2];

  o[43] = allz ? 1.0f : 0.0f;
  o[44] = aacc;
  o[45] = dacc;
  o[46] = 0.0f;
  o[47] = 0.0f;
}

// ---------------- scalar MLP fallback (only if WMMA builtin is absent) -------
__device__ __forceinline__ float mlp_logit(float x, const float* q) {
  float l0[3], l1[3], l2[3];
#pragma unroll
  for (int j = 0; j < 3; ++j) {
    float v = fmaf(q[0 + j], x, q[3 + j]);
    v = fmaf(q[34 + j], fast_tanh(v), v);
    l0[j] = v;
  }
#pragma unroll
  for (int i = 0; i < 3; ++i) {
    float v = q[15 + i];
#pragma unroll
    for (int j = 0; j < 3; ++j) v = fmaf(q[6 + i * 3 + j], l0[j], v);
    v = fmaf(q[37 + i], fast_tanh(v), v);
    l1[i] = v;
  }
#pragma unroll
  for (int i = 0; i < 3; ++i) {
    float v = q[27 + i];
#pragma unroll
    for (int j = 0; j < 3; ++j) v = fmaf(q[18 + i * 3 + j], l1[j], v);
    v = fmaf(q[40 + i], fast_tanh(v), v);
    l2[i] = v;
  }
  float v = q[33];
#pragma unroll
  for (int j = 0; j < 3; ++j) v = fmaf(q[30 + j], l2[j], v);
  return v;
}

// ---------------- async-tensor wait helpers ----------------
#if __has_builtin(__builtin_amdgcn_s_wait_asynccnt)
#define WAIT_ASYNC_1() __builtin_amdgcn_s_wait_asynccnt(1)
#define WAIT_ASYNC_0() __builtin_amdgcn_s_wait_asynccnt(0)
#else
#define WAIT_ASYNC_1() asm volatile("s_wait_asynccnt 0x1" ::: "memory")
#define WAIT_ASYNC_0() asm volatile("s_wait_asynccnt 0x0" ::: "memory")
#endif

__device__ __forceinline__ void async_load_b128(unsigned lds_addr, unsigned voff,
                                                unsigned long long gbase) {
  // GVS mode: mem = SADDR(64) + VADDR(32) ; lds dest addr comes from VDST vgpr
  asm volatile("global_load_async_to_lds_b128 %0, %1, %2"
               :: "v"(lds_addr), "v"(voff), "s"(gbase)
               : "memory");
}

// ---------------- WMMA helpers for the general (gated) path ------------------
// A-layout for V_WMMA_F32_16X16X4_F32: 16x4 (MxK) f32 in 2 VGPRs.
//   lanes 0-15 (M=lane):  A.x = W[M][K=0], A.y = W[M][K=1]
//   lanes 16-31 (M=lane-16): A.x = W[M][K=2], A.y = W[M][K=3] (=0 here)
// We duplicate W's rows 0..2 into rows 8..10 so D comes back with rows 0..2
// broadcast into BOTH lane halves (D vgpr r holds M=r | M=8+r) -> the next
// layer's B operand is a pure lane-local select, no cross-lane moves.
__device__ __forceinline__ v2f build_A3x3(const float* q, int base, int lane) {
  const int half = lane >> 4;
  const int m = lane & 15;
  const int rr = (m < 3) ? m : ((m >= 8 && m < 11) ? (m - 8) : -1);
  float a0 = 0.0f, a1 = 0.0f;
  if (rr >= 0) {
    a0 = q[base + rr * 3 + (half ? 2 : 0)];     // K=0 or K=2
    a1 = half ? 0.0f : q[base + rr * 3 + 1];    // K=1 (K=3 is zero)
  }
  v2f a; a.x = a0; a.y = a1;
  return a;
}

// ---------------- main streaming kernel ----------------
__global__ __launch_bounds__(256)
void fp_main(const float* __restrict__ z, const float* __restrict__ P,
             float* __restrict__ out) {
  __shared__ v4f smem[2][256];      // double-buffered 4KB input tiles
  __shared__ float stage[2048];     // general path: staged logit inputs (wave-private regions)
  __shared__ float resbuf[2048];    // general path: sigmoid results
  const int slice = blockIdx.x;     // b*C + c
  const int c = slice % CC;
  const int tid = threadIdx.x;

  // Per-channel params: block-uniform -> scalar (SMEM) loads
  float q[46];
  {
    const float* p = P + c * PSTRIDE;
#pragma unroll
    for (int k = 0; k < 46; ++k) q[k] = p[k];
  }
  const bool fastpath = (q[43] != 0.0f);
  const float a = q[44], d = q[45];
  const float du = fmaf(0.5f, a, d);
  const float dl = fmaf(-0.5f, a, d);

  const unsigned long long gbase =
      (unsigned long long)(uintptr_t)(const void*)(z + (size_t)slice * HW);
  v4f* __restrict__ outz = (v4f*)out + (size_t)slice * (HW / 4);
  v4f* __restrict__ outl =
      (v4f*)out + (size_t)(NTOT / 4) + (size_t)slice * (HW / 4);

  // Low 32 bits of a generic LDS pointer == LDS byte offset (flat-LDS truncation)
  unsigned lds0 = (unsigned)(uintptr_t)&smem[0][tid];
  unsigned lds1 = (unsigned)(uintptr_t)&smem[1][tid];

  // Each wave's async load covers exactly the 512B its own lanes consume,
  // so per-wave ASYNCcnt waits suffice (no block barriers needed).
  async_load_b128(lds0, tid * 16u, gbase);
  async_load_b128(lds1, 4096u + tid * 16u, gbase);

#pragma unroll
  for (int t = 0; t < NT; ++t) {
    if (t < NT - 1) { WAIT_ASYNC_1(); } else { WAIT_ASYNC_0(); }

    v4f x = smem[t & 1][tid];

    if (t + 2 < NT) {
      // Our ds reads must retire before TDM overwrites this buffer (WAR)
      asm volatile("s_wait_dscnt 0x0" ::: "memory");
      async_load_b128((t & 1) ? lds1 : lds0,
                      (unsigned)(t + 2) * 4096u + tid * 16u, gbase);
    }

    float zi[4] = {x.x, x.y, x.z, x.w};
    float zh[4], lk[4];
    if (fastpath) {
#pragma unroll
      for (int k = 0; k < 4; ++k) {
        float r = rintf(zi[k]);                   // round-half-even, matches jnp.round
        zh[k] = r;
        float u = fast_sigmoid(fmaf(a, r, du));
        float l = fast_sigmoid(fmaf(a, r, dl));
        lk[k] = fmaxf(u - l, 1e-9f);
      }
    } else {
#if __has_builtin(__builtin_amdgcn_wmma_f32_16x16x4_f32)
      // ---- WMMA general path: 16 logit evals per batch, layers 1&2 as
      // V_WMMA_F32_16X16X4_F32 (f32, K=4>=3), zero cross-lane shuffles. ----
      const int lane = tid & 31;
      const int wavebase = (tid >> 5) * 256;   // this wave's private region
      const int half = lane >> 4;
      const int col = lane & 15;

      // Stage 8 inputs per thread (upper/lower for 4 pixels) in wave-private LDS
#pragma unroll
      for (int k = 0; k < 4; ++k) {
        float r = rintf(zi[k]);
        zh[k] = r;
        stage[tid * 8 + k * 2 + 0] = r + 0.5f;
        stage[tid * 8 + k * 2 + 1] = r - 0.5f;
      }

      const v2f A1 = build_A3x3(q, 6, lane);    // W1 rows duplicated at M=8..10
      const v2f A2 = build_A3x3(q, 18, lane);   // W2 likewise
      v8f c1 = {q[15], q[16], q[17], 0.f, 0.f, 0.f, 0.f, 0.f};  // b1 in C rows 0..2 (and 8..10 via dup rows? rows 8..10 get same bias rows 0..2 values broadcast per-VGPR)
      v8f c2 = {q[27], q[28], q[29], 0.f, 0.f, 0.f, 0.f, 0.f};  // b2

      // 16 batches of 16 evals = this wave's 256 staged values
#pragma unroll 4
      for (int bb = 0; bb < 16; ++bb) {
        float xv = stage[wavebase + bb * 16 + col];  // both halves read same col
        // layer 0 (rank-1): every lane computes the full 3-vector locally
        float l00 = fmaf(q[0], xv, q[3]);  l00 = fmaf(q[34], fast_tanh(l00), l00);
        float l01 = fmaf(q[1], xv, q[4]);  l01 = fmaf(q[35], fast_tanh(l01), l01);
        float l02 = fmaf(q[2], xv, q[5]);  l02 = fmaf(q[36], fast_tanh(l02), l02);
        // B(4x16): vgpr0 = {K0 | K2}, vgpr1 = {K1 | 0}
        v2f B1v; B1v.x = half ? l02 : l00; B1v.y = half ? 0.0f : l01;
        v8f D1 = __builtin_amdgcn_wmma_f32_16x16x4_f32(
            false, A1, false, B1v, (short)0, c1, false, false);
        // rows 0..2 broadcast in both halves; apply tanh gating
        float u0 = fmaf(q[37], fast_tanh(D1[0]), D1[0]);
        float u1 = fmaf(q[38], fast_tanh(D1[1]), D1[1]);
        float u2 = fmaf(q[39], fast_tanh(D1[2]), D1[2]);
        v2f B2v; B2v.x = half ? u2 : u0; B2v.y = half ? 0.0f : u1;
        v8f D2 = __builtin_amdgcn_wmma_f32_16x16x4_f32(
            false, A2, false, B2v, (short)0, c2, false, false);
        float t0 = fmaf(q[40], fast_tanh(D2[0]), D2[0]);
        float t1 = fmaf(q[41], fast_tanh(D2[1]), D2[1]);
        float t2 = fmaf(q[42], fast_tanh(D2[2]), D2[2]);
        // layer 3 (rank-1) + sigmoid
        float lg = q[33];
        lg = fmaf(q[30], t0, lg);
        lg = fmaf(q[31], t1, lg);
        lg = fmaf(q[32], t2, lg);
        float sg = fast_sigmoid(lg);
        if (half == 0) resbuf[wavebase + bb * 16 + col] = sg;
      }

      // Read back this thread's 8 results (wave-private; DScnt ordering only)
#pragma unroll
      for (int k = 0; k < 4; ++k) {
        float u = resbuf[tid * 8 + k * 2 + 0];
        float l = resbuf[tid * 8 + k * 2 + 1];
        lk[k] = fmaxf(u - l, 1e-9f);
      }
#else
#pragma unroll
      for (int k = 0; k < 4; ++k) {
        float r = rintf(zi[k]);
        zh[k] = r;
        float u = fast_sigmoid(mlp_logit(r + 0.5f, q));
        float l = fast_sigmoid(mlp_logit(r - 0.5f, q));
        lk[k] = fmaxf(u - l, 1e-9f);
      }
#endif
    }

    const int gi = t * 256 + tid;
    v4f vz = {zh[0], zh[1], zh[2], zh[3]};
    v4f vl = {lk[0], lk[1], lk[2], lk[3]};
    __builtin_nontemporal_store(vz, &outz[gi]);
    __builtin_nontemporal_store(vl, &outl[gi]);
  }
}

extern "C" void kernel_launch(void* const* d_in, const int* in_sizes, int n_in,
                              void* d_out, int out_size, void* d_ws, size_t ws_size,
                              hipStream_t stream) {
  const float* z  = (const float*)d_in[0];
  const float* m0 = (const float*)d_in[1];
  const float* m1 = (const float*)d_in[2];
  const float* m2 = (const float*)d_in[3];
  const float* m3 = (const float*)d_in[4];
  const float* b0 = (const float*)d_in[5];
  const float* b1 = (const float*)d_in[6];
  const float* b2 = (const float*)d_in[7];
  const float* b3 = (const float*)d_in[8];
  const float* f0 = (const float*)d_in[9];
  const float* f1 = (const float*)d_in[10];
  const float* f2 = (const float*)d_in[11];

  float* P = (float*)d_ws;  // 192 * 48 * 4 = 36,864 bytes

  fp_prep<<<1, 256, 0, stream>>>(m0, m1, m2, m3, b0, b1, b2, b3, f0, f1, f2, P);
  fp_main<<<NSL, 256, 0, stream>>>(z, P, (float*)d_out);
}